// ReconEncoder_87402584473723
// MI455X (gfx1250) — compile-verified
//
#include <hip/hip_runtime.h>

static constexpr int NN = 100000;   // nodes
static constexpr int NE = 3200000;  // edges
static constexpr int DF = 256;      // in feat
static constexpr int DH = 256;      // hidden
static constexpr int DL = 128;      // latent

typedef __attribute__((ext_vector_type(2))) float v2f;
typedef __attribute__((ext_vector_type(8))) float v8f;

// ---------------- degree / norm ----------------
__global__ void k_init_deg(float* deg) {
    int i = blockIdx.x * blockDim.x + threadIdx.x;
    if (i < NN) deg[i] = 1.0f;  // self loop contributes 1
}

__global__ void k_count_deg(const long long* __restrict__ dst, float* __restrict__ deg) {
    int i = blockIdx.x * blockDim.x + threadIdx.x;
    if (i < NE) atomicAdd(&deg[(int)dst[i]], 1.0f);
}

__global__ void k_dinv(float* deg) {
    int i = blockIdx.x * blockDim.x + threadIdx.x;
    if (i < NN) deg[i] = rsqrtf(deg[i]);  // deg >= 1 always (self loop)
}

__global__ void k_bias_sum(const float* __restrict__ b1, const float* __restrict__ blin,
                           float* __restrict__ bsum) {
    int i = threadIdx.x;
    if (i < DH) bsum[i] = b1[i] + blin[i];
}

// ---------------- fp32 WMMA GEMM ----------------
// C[M x N] = A[M x K] @ B[K x N] (+ bias[N]).
// K, N compile-time so all loads use immediate offsets and the K loop
// fully unrolls. Each wave computes TWO 16x16 C tiles (32 rows) sharing the
// same B fragment -> two independent WMMA accumulate chains (ILP=2, half the
// B traffic). blockDim.x = 128 (4 waves); grid = (M/32, N/64).
// ISA 16x4 f32 A layout: lane = (K>=2)*16 + M, vgpr v holds K = 2*(lane>>4)+v.
// C/D layout: vgpr r -> M = r + (lane>>4)*8, N = lane&15.
template <int K, int N, bool HASBIAS>
__global__ void k_gemm_wmma(const float* __restrict__ A, const float* __restrict__ B,
                            const float* __restrict__ bias, float* __restrict__ C) {
    const int wave  = threadIdx.x >> 5;
    const int lane  = threadIdx.x & 31;
    const int mn    = lane & 15;
    const int half  = lane >> 4;
    const int tileM = blockIdx.x * 2;           // two M tiles per wave
    const int tileN = blockIdx.y * 4 + wave;

    const float* Arow0 = A + (size_t)(tileM * 16 + mn) * K + half * 2;
    const float* Arow1 = Arow0 + (size_t)16 * K;
    const int    col   = tileN * 16 + mn;
    const float* Bp    = B + (size_t)(half * 2) * N + col;

    v8f acc0 = {};
    v8f acc1 = {};
#pragma unroll
    for (int kk = 0; kk < K; kk += 4) {
        const v2f a0 = *reinterpret_cast<const v2f*>(Arow0 + kk);  // 8B aligned
        const v2f a1 = *reinterpret_cast<const v2f*>(Arow1 + kk);
        v2f b;
        b.x = Bp[(size_t)kk * N];
        b.y = Bp[(size_t)(kk + 1) * N];
        acc0 = __builtin_amdgcn_wmma_f32_16x16x4_f32(
            false, a0, false, b, (short)0, acc0, false, false);
        acc1 = __builtin_amdgcn_wmma_f32_16x16x4_f32(
            false, a1, false, b, (short)0, acc1, false, false);
    }

    const float bv = HASBIAS ? bias[col] : 0.0f;
    float* C0 = C + (size_t)(tileM * 16 + half * 8) * N + col;
#pragma unroll
    for (int r = 0; r < 8; ++r) {
        C0[(size_t)r * N]        = acc0[r] + bv;   // tile 0: rows +0..7 (+8 via half)
        C0[(size_t)(16 + r) * N] = acc1[r] + bv;   // tile 1: rows +16..23 (+8 via half)
    }
}

// ---------------- edge scatter (message passing) ----------------
// out[dst] += msg[src] * dinv[src]*dinv[dst]; F floats per node row.
// Each thread owns one float4; F/4 threads per edge; 256-thread blocks.
template <int F>
__global__ void k_scatter(const long long* __restrict__ srcs, const long long* __restrict__ dsts,
                          const float* __restrict__ dinv, const float* __restrict__ msg,
                          float* __restrict__ out) {
    constexpr int TPE = F / 4;        // threads per edge
    constexpr int EPB = 256 / TPE;    // edges per block
    const int tid = threadIdx.x;
    const long e  = (long)blockIdx.x * EPB + tid / TPE;
    if (e >= NE) return;
    const int s = (int)srcs[e];
    const int d = (int)dsts[e];
    const float nrm = dinv[s] * dinv[d];
    const int f = (tid % TPE) * 4;
    const float4 m = *reinterpret_cast<const float4*>(msg + (size_t)s * F + f);
    float* o = out + (size_t)d * F + f;
    atomicAdd(o + 0, m.x * nrm);
    atomicAdd(o + 1, m.y * nrm);
    atomicAdd(o + 2, m.z * nrm);
    atomicAdd(o + 3, m.w * nrm);
}

// agg += t1 * dinv^2 (self loop), then ReLU -> z (in place in agg)
__global__ void k_selfloop_relu(const float* __restrict__ t1, const float* __restrict__ dinv,
                                float* __restrict__ agg) {
    const long i = (long)blockIdx.x * blockDim.x + threadIdx.x;
    if (i >= (long)NN * DH) return;
    const int   node = (int)(i / DH);
    const float di   = dinv[node];
    const float v    = agg[i] + t1[i] * di * di;
    agg[i] = v > 0.0f ? v : 0.0f;
}

// out = b2 + t2 * dinv^2 (bias + self loop term of layer 2)
__global__ void k_out_init(const float* __restrict__ t2, const float* __restrict__ dinv,
                           const float* __restrict__ b2, float* __restrict__ out) {
    const long i = (long)blockIdx.x * blockDim.x + threadIdx.x;
    if (i >= (long)NN * DL) return;
    const int   node = (int)(i / DL);
    const int   f    = (int)(i % DL);
    const float di   = dinv[node];
    out[i] = b2[f] + t2[i] * di * di;
}

extern "C" void kernel_launch(void* const* d_in, const int* in_sizes, int n_in,
                              void* d_out, int out_size, void* d_ws, size_t ws_size,
                              hipStream_t stream) {
    const float*     x    = (const float*)d_in[0];
    const long long* ei   = (const long long*)d_in[1];  // int64 [2, NE]
    const float*     W1   = (const float*)d_in[2];
    const float*     b1   = (const float*)d_in[3];
    const float*     Wlin = (const float*)d_in[4];
    const float*     blin = (const float*)d_in[5];
    const float*     W2   = (const float*)d_in[6];
    const float*     b2   = (const float*)d_in[7];
    float*           out  = (float*)d_out;

    const long long* esrc = ei;
    const long long* edst = ei + NE;

    // workspace layout (floats): dinv | bsum | t1 (also t2) | agg (also z)
    float* ws   = (float*)d_ws;
    float* dinv = ws;
    float* bsum = ws + (((size_t)NN + 255) & ~(size_t)255);
    float* t1   = bsum + 256;
    float* agg  = t1 + (size_t)NN * DH;

    // 1) symmetric normalization
    k_init_deg <<<(NN + 255) / 256, 256, 0, stream>>>(dinv);
    k_count_deg<<<(NE + 255) / 256, 256, 0, stream>>>(edst, dinv);
    k_dinv     <<<(NN + 255) / 256, 256, 0, stream>>>(dinv);
    k_bias_sum <<<1, 256, 0, stream>>>(b1, blin, bsum);

    const dim3 blk(128);
    // 2) t1 = x @ W1 ; agg = x @ Wlin + (b1+blin)
    k_gemm_wmma<DF, DH, false><<<dim3(NN / 32, DH / 64), blk, 0, stream>>>(x, W1, nullptr, t1);
    k_gemm_wmma<DF, DH, true ><<<dim3(NN / 32, DH / 64), blk, 0, stream>>>(x, Wlin, bsum, agg);

    // 3) agg[dst] += t1[src] * norm   (E edges, 4 edges / 256-thread block)
    k_scatter<DH><<<(NE + 3) / 4, 256, 0, stream>>>(esrc, edst, dinv, t1, agg);

    // 4) z = relu(agg + t1*dinv^2)  (in place)
    k_selfloop_relu<<<(unsigned)(((long)NN * DH + 255) / 256), 256, 0, stream>>>(t1, dinv, agg);

    // 5) t2 = z @ W2 (reuse t1 buffer)
    float* t2 = t1;
    k_gemm_wmma<DH, DL, false><<<dim3(NN / 32, DL / 64), blk, 0, stream>>>(agg, W2, nullptr, t2);

    // 6) out = b2 + t2*dinv^2, then out[dst] += t2[src]*norm
    k_out_init<<<(unsigned)(((long)NN * DL + 255) / 256), 256, 0, stream>>>(t2, dinv, b2, out);
    k_scatter<DL><<<(NE + 7) / 8, 256, 0, stream>>>(esrc, edst, dinv, t2, out);
}